// Attention_16338055594502
// MI455X (gfx1250) — compile-verified
//
#include <hip/hip_runtime.h>
#include <hip/hip_bf16.h>
#include <math.h>

typedef __attribute__((ext_vector_type(2))) float v2f;
typedef __attribute__((ext_vector_type(8))) float v8f;

#define C_DIM 128
#define H_DIM 8

// ---------------------------------------------------------------------------
// Kernel 1: per-node projection  S[n][0:8] = x[n]@Wr,  S[n][8:16] = x[n]@Wc
// One wave32 per 16-node tile; V_WMMA_F32_16X16X4_F32 over K=128 (32 steps).
// W is (2C, H) row-major; our B matrix is 128x16 with
//   B[k][n] = (n < 8) ? W[k][n] : W[C + k][n - 8]
// ---------------------------------------------------------------------------
__global__ __launch_bounds__(32) void node_proj_kernel(
    const float* __restrict__ x,   // (N, 128)
    const float* __restrict__ W,   // (256, 8)
    float* __restrict__ S,         // (N, 16)  workspace
    int N)
{
    const int lane = threadIdx.x & 31;
    const int half = lane >> 4;        // 0: K pair {0,1}, 1: K pair {2,3}
    const int l16  = lane & 15;
    const int tile = blockIdx.x;

    // A-matrix row for this lane (clamp so EXEC stays all-1s for WMMA)
    int m = tile * 16 + l16;
    if (m >= N) m = N - 1;
    const float* xrow = x + (size_t)m * C_DIM;

    // B column for this lane
    const int n = l16;
    const float* Wcol = (n < H_DIM) ? (W + n) : (W + (size_t)C_DIM * H_DIM + (n - H_DIM));

    v8f acc = {};
    #pragma unroll
    for (int k0 = 0; k0 < C_DIM; k0 += 4) {
        const int k = k0 + 2 * half;
        // A 16x4 layout: lanes 0-15 -> K = k0,k0+1 ; lanes 16-31 -> K = k0+2,k0+3
        v2f a;
        a.x = xrow[k + 0];
        a.y = xrow[k + 1];
        // B 4x16 layout: VGPR0 = rows k0 / k0+2, VGPR1 = rows k0+1 / k0+3
        v2f bm;
        bm.x = Wcol[(size_t)(k + 0) * H_DIM];
        bm.y = Wcol[(size_t)(k + 1) * H_DIM];
        acc = __builtin_amdgcn_wmma_f32_16x16x4_f32(
            /*neg_a=*/false, a, /*neg_b=*/false, bm,
            /*c_mod=*/(short)0, acc, /*reuse_a=*/false, /*reuse_b=*/false);
    }

    // D layout: VGPR v, lane -> row M = v + 8*half, col N = l16
    #pragma unroll
    for (int v = 0; v < 8; ++v) {
        const int mm = tile * 16 + v + 8 * half;
        if (mm < N) S[(size_t)mm * 16 + l16] = acc[v];
    }
}

// ---------------------------------------------------------------------------
// Kernel 2: streaming edge pass.
//   alpha[e][h] = (row==col) ? 1 : edge_attr[e] * sigmoid(S[row][h] + S[col][8+h] + b[h])
// S (640 KB) is L2-resident; this kernel is HBM-bandwidth bound on
// edge_index reads + alpha writes.
// ---------------------------------------------------------------------------
__global__ __launch_bounds__(256) void edge_score_kernel(
    const long long* __restrict__ ei,   // (2, E) int64
    const float* __restrict__ ea,       // (E,)
    const float* __restrict__ S,        // (N, 16)
    const float* __restrict__ b,        // (8,)
    float* __restrict__ alpha,          // (E, 8)
    int E)
{
    const int e = blockIdx.x * 256 + (int)threadIdx.x;
    if (e >= E) return;

    // Prefetch upcoming edge data into cache (global_prefetch_b8)
    if (e + 8192 < E) {
        __builtin_prefetch(&ei[e + 8192], 0, 1);
        __builtin_prefetch(&ei[(size_t)E + e + 8192], 0, 1);
        __builtin_prefetch(&ea[e + 8192], 0, 1);
    }

    const long long r = ei[e];
    const long long c = ei[(size_t)E + e];
    const float attr  = ea[e];
    const bool self   = (r == c);

    const float4* sr = (const float4*)(S + (size_t)r * 16);      // S[row][0:8]
    const float4* sc = (const float4*)(S + (size_t)c * 16 + 8);  // S[col][8:16]
    const float4 sr0 = sr[0], sr1 = sr[1];
    const float4 sc0 = sc[0], sc1 = sc[1];
    const float4 b0 = ((const float4*)b)[0];
    const float4 b1 = ((const float4*)b)[1];

    float z[8];
    z[0] = sr0.x + sc0.x + b0.x;  z[1] = sr0.y + sc0.y + b0.y;
    z[2] = sr0.z + sc0.z + b0.z;  z[3] = sr0.w + sc0.w + b0.w;
    z[4] = sr1.x + sc1.x + b1.x;  z[5] = sr1.y + sc1.y + b1.y;
    z[6] = sr1.z + sc1.z + b1.z;  z[7] = sr1.w + sc1.w + b1.w;

    float o[8];
    #pragma unroll
    for (int h = 0; h < 8; ++h) {
        const float sig = 1.0f / (1.0f + __expf(-z[h]));   // v_exp_f32 + v_rcp_f32
        o[h] = self ? 1.0f : attr * sig;
    }

    float4* dst = (float4*)(alpha + (size_t)e * 8);
    dst[0] = make_float4(o[0], o[1], o[2], o[3]);
    dst[1] = make_float4(o[4], o[5], o[6], o[7]);
}

// ---------------------------------------------------------------------------
// Kernel 3: passthrough copy of edge_index (2*E int64 == E int4) to output tail.
// ---------------------------------------------------------------------------
__global__ __launch_bounds__(256) void copy_index_kernel(
    const int4* __restrict__ src, int4* __restrict__ dst, int n)
{
    const int i = blockIdx.x * 256 + (int)threadIdx.x;
    if (i < n) dst[i] = src[i];
}

extern "C" void kernel_launch(void* const* d_in, const int* in_sizes, int n_in,
                              void* d_out, int out_size, void* d_ws, size_t ws_size,
                              hipStream_t stream) {
    const float*     x  = (const float*)d_in[0];      // (N, 128)
    const long long* ei = (const long long*)d_in[1];  // (2, E) int64
    const float*     ea = (const float*)d_in[2];      // (E,)
    const float*     W  = (const float*)d_in[3];      // (256, 8)
    const float*     b  = (const float*)d_in[4];      // (8,)

    const int N = in_sizes[0] / C_DIM;
    const int E = in_sizes[2];

    float* S     = (float*)d_ws;                      // (N, 16) = 640 KB
    float* alpha = (float*)d_out;                     // (E, 8) f32, then 2*E i64

    const int tiles = (N + 15) / 16;
    node_proj_kernel<<<tiles, 32, 0, stream>>>(x, W, S, N);

    edge_score_kernel<<<(E + 255) / 256, 256, 0, stream>>>(ei, ea, S, b, alpha, E);

    // edge_index passthrough: 2*E int64 = E int4, placed after E*8 floats
    int4* idx_dst = (int4*)(alpha + (size_t)E * 8);
    copy_index_kernel<<<(E + 255) / 256, 256, 0, stream>>>((const int4*)ei, idx_dst, E);
}